// LlamaAttention_30167850287826
// MI455X (gfx1250) — compile-verified
//
#include <hip/hip_runtime.h>
#include <stdint.h>

// ---------------- problem constants ----------------
#define BDIM   8
#define HDIM   32
#define KVH    8
#define GRP    4        // H / KVH
#define DDIM   128
#define BLK    16
#define MAXS   4096
#define NBT    (MAXS / BLK)   // 256 block-table entries per sequence

// split-S configuration: 8 chunks x 4 waves -> 512 workgroups, NPART=32
#define NCHUNK 8
#define CHUNK  (MAXS / NCHUNK)        // 512 positions per workgroup
#define NWAVES 4
#define TPW    (CHUNK / 16 / NWAVES)  // 8 tiles (of 16 tokens) per wave
#define NPART  (NCHUNK * NWAVES)      // 32 partials per (b,kvh)
#define PSTRIDE 520                   // floats per partial: 4 m + 4 l + 4*128 acc

// LDS layout (floats):
//   [Qpad 16x128][per-wave: K0 | V0 | K1 | V1 (each 16x128) | W 16x16]
#define QLDS_F   2048
#define TILE_F   2048
#define WAVE_F   (4 * TILE_F + 256)    // 8448
#define SMEM_F   (QLDS_F + NWAVES * WAVE_F)
#define SMEM_B   (SMEM_F * 4)          // 143360 bytes -> 2 WGs per WGP

typedef float v2f __attribute__((ext_vector_type(2)));
typedef float v8f __attribute__((ext_vector_type(8)));
typedef unsigned int v4u __attribute__((ext_vector_type(4)));
typedef int v8i __attribute__((ext_vector_type(8)));
typedef int v4i __attribute__((ext_vector_type(4)));

// ---- TDM: 2-D tile load, 16 rows x 128 f32, row stride KVH*D elems ----
// D# packing per CDNA5 ISA (groups 0/1, 2D mode).
// clang-23 6-arg builtin form.
__device__ __forceinline__ void tdm_load_tile16x128(uint32_t ldsByteOff,
                                                    const float* gptr) {
  uint64_t ga = (uint64_t)(uintptr_t)gptr;
  v4u g0;
  g0[0] = 1u;                                    // count=1 (valid), user mode
  g0[1] = ldsByteOff;                            // lds_addr
  g0[2] = (uint32_t)ga;                          // global_addr[31:0]
  g0[3] = (uint32_t)((ga >> 32) & 0x1FFFFFFull)  // global_addr[56:32]
          | (2u << 30);                          // type=2 ("image")
  v8i g1;
  g1[0] = (int)(2u << 16);        // workgroup_mask=0, data_size=2 (4B)
  g1[1] = (int)(128u << 16);      // tensor_dim0 = 128 (bits 79:48, low 16)
  g1[2] = (int)(16u << 16);       // tensor_dim0 hi=0 | tensor_dim1=16 (low16)
  g1[3] = (int)(128u << 16);      // tensor_dim1 hi=0 | tile_dim0=128
  g1[4] = 16;                     // tile_dim1=16, tile_dim2=0
  g1[5] = (int)(KVH * DDIM);      // tensor_dim0_stride = 1024 elems (low 32)
  g1[6] = 0;                      // stride hi | dim1_stride low
  g1[7] = 0;
  v4i z4; z4[0] = z4[1] = z4[2] = z4[3] = 0;      // groups 2/3 unused (2-D)
  v8i z8;
#pragma unroll
  for (int i = 0; i < 8; ++i) z8[i] = 0;
  __builtin_amdgcn_tensor_load_to_lds(g0, g1, z4, z4, z8, 0);
}

__device__ __forceinline__ v8f zero8() {
  v8f r;
#pragma unroll
  for (int i = 0; i < 8; ++i) r[i] = 0.f;
  return r;
}

// ---------------- kernel 1: RoPE for Q and current K ----------------
__global__ void rope_kernel(const float* __restrict__ Q,
                            const float* __restrict__ K,
                            const float* __restrict__ cosb,
                            const float* __restrict__ sinb,
                            float* __restrict__ qr,
                            float* __restrict__ kq) {
  int i = blockIdx.x * 256 + threadIdx.x;
  if (i < BDIM * HDIM * DDIM) {
    int b = i >> 12;            // / (H*D)
    int d = i & (DDIM - 1);
    float x  = Q[i];
    float xr = (d < DDIM / 2) ? -Q[i + DDIM / 2] : Q[i - DDIM / 2];
    qr[i] = x * cosb[(b << 7) + d] + xr * sinb[(b << 7) + d];
  } else if (i < BDIM * HDIM * DDIM + BDIM * KVH * DDIM) {
    int j = i - BDIM * HDIM * DDIM;
    int b = j >> 10;            // / (KVH*D)
    int d = j & (DDIM - 1);
    float x  = K[j];
    float xr = (d < DDIM / 2) ? -K[j + DDIM / 2] : K[j - DDIM / 2];
    kq[j] = x * cosb[(b << 7) + d] + xr * sinb[(b << 7) + d];
  }
}

// ---------------- kernel 2: streaming flash-decode partials ----------------
__global__ void attn_partial_kernel(const float* __restrict__ qr,
                                    const float* __restrict__ kq,
                                    const float* __restrict__ Vcur,
                                    const float* __restrict__ Kc,
                                    const float* __restrict__ Vc,
                                    const int*   __restrict__ lens,
                                    const float* __restrict__ maskp,
                                    const int*   __restrict__ btab,
                                    float* __restrict__ part) {
  const int b = blockIdx.z, kvh = blockIdx.y, chunk = blockIdx.x;
  const int tid  = threadIdx.x;
  const int lane = tid & 31;
  const int wave = __builtin_amdgcn_readfirstlane(tid >> 5);

  extern __shared__ float smem[];
  float* qlds  = smem;
  float* wbase = smem + QLDS_F + wave * WAVE_F;   // K0 V0 K1 V1 W
  float* wlds  = wbase + 4 * TILE_F;
  const uint32_t bufOff0 = (uint32_t)((QLDS_F + wave * WAVE_F) * 4);

  const int   len = lens[b];
  const float sc  = 0.08838834764831845f * maskp[0];  // D^-0.5 * mask

  // Q tile: rows 0..3 = heads g*KVH+kvh (group-major GQA), rows 4..15 = 0 pad
  for (int idx = tid; idx < QLDS_F; idx += NWAVES * 32) {
    int r = idx >> 7, d = idx & 127;
    qlds[idx] = (r < GRP) ? qr[((b * HDIM + r * KVH + kvh) << 7) + d] : 0.f;
  }
  // zero pad rows 4..15 of this wave's W tile (never rewritten)
#pragma unroll
  for (int r = 0; r < 6; ++r) wlds[64 + r * 32 + lane] = 0.f;
  __syncthreads();

  const int row16 = lane & 15;
  const int kh2   = (lane >> 4) << 1;   // 2*(lane/16): WMMA frag K offset

  float mrow[GRP], lrow[GRP];
  v8f   acc[8];
#pragma unroll
  for (int g = 0; g < GRP; ++g) { mrow[g] = -__builtin_inff(); lrow[g] = 0.f; }
#pragma unroll
  for (int n = 0; n < 8; ++n) acc[n] = zero8();

  const int plast = len - 1;
  const int wstart = chunk * CHUNK + wave * (TPW * 16);   // uniform
  const int remain = len - wstart;
  const int ntiles = remain <= 0 ? 0
                   : (remain >= TPW * 16 ? TPW : ((remain + 15) >> 4));

  // prime the ping-pong pipeline: issue tile 0 into buffer 0
  if (ntiles > 0) {
    const int bt0 =
        __builtin_amdgcn_readfirstlane(btab[b * NBT + (wstart >> 4)]);
    const size_t g0 = ((size_t)bt0 * BLK) * (KVH * DDIM) + (size_t)kvh * DDIM;
    tdm_load_tile16x128(bufOff0, Kc + g0);
    tdm_load_tile16x128(bufOff0 + (uint32_t)(TILE_F * 4), Vc + g0);
  }

  for (int t = 0; t < ntiles; ++t) {
    const int base = wstart + t * 16;
    const int cur  = t & 1;

    // prefetch tile t+1 into the other buffer, then wait for tile t only
    if (t + 1 < ntiles) {
      const int nb = wstart + (t + 1) * 16;
      const int bt =
          __builtin_amdgcn_readfirstlane(btab[b * NBT + (nb >> 4)]);
      const size_t go = ((size_t)bt * BLK) * (KVH * DDIM) + (size_t)kvh * DDIM;
      const uint32_t off = bufOff0 + (uint32_t)(((t + 1) & 1) * 2 * TILE_F * 4);
      tdm_load_tile16x128(off, Kc + go);
      tdm_load_tile16x128(off + (uint32_t)(TILE_F * 4), Vc + go);
      __builtin_amdgcn_s_wait_tensorcnt(2);   // TDM in-order: tile t done
    } else {
      __builtin_amdgcn_s_wait_tensorcnt(0);
    }
    __asm__ __volatile__("" ::: "memory");

    float* klds = wbase + cur * 2 * TILE_F;
    float* vlds = klds + TILE_F;

    if ((plast >> 4) == (base >> 4)) {            // inject current token row
      const int r = plast & 15;
      ((float4*)(klds + r * DDIM))[lane] =
          ((const float4*)(kq + (b * KVH + kvh) * DDIM))[lane];
      ((float4*)(vlds + r * DDIM))[lane] =
          ((const float4*)(Vcur + (b * KVH + kvh) * DDIM))[lane];
    }

    // ---- scores = Qpad(16x128) x Ktile^T(128x16): 32 x wmma f32 16x16x4 ----
    v8f s = zero8();
#pragma unroll
    for (int kk = 0; kk < 32; ++kk) {
      const int d0 = (kk << 2) + kh2;
      v2f a  = *(const v2f*)(qlds + row16 * DDIM + d0);
      v2f kb = *(const v2f*)(klds + row16 * DDIM + d0);
      s = __builtin_amdgcn_wmma_f32_16x16x4_f32(false, a, false, kb,
                                                (short)0, s, false, false);
    }

    // ---- online softmax over this 16-token tile (rows 0..3 real) ----
    const int  pj    = base + row16;
    const bool valid = pj < len;
#pragma unroll
    for (int g = 0; g < GRP; ++g) {
      float v = valid ? s[g] * sc : -__builtin_inff();
      float tm = v;
#pragma unroll
      for (int mm = 8; mm >= 1; mm >>= 1)
        tm = fmaxf(tm, __shfl_xor(tm, mm, 32));
      const float mnew = fmaxf(mrow[g], tm);
      const float f    = __expf(mrow[g] - mnew);
      float w = valid ? __expf(v - mnew) : 0.f;
      float rs = w;
#pragma unroll
      for (int mm = 8; mm >= 1; mm >>= 1) rs += __shfl_xor(rs, mm, 32);
      lrow[g] = lrow[g] * f + rs;
      mrow[g] = mnew;
#pragma unroll
      for (int n = 0; n < 8; ++n) acc[n][g] *= f;
      if (lane < 16) wlds[g * 16 + lane] = w;     // W row g (rows 4..15 = 0)
    }

    // ---- acc += W(16x16) x Vtile(16x128): 8 N-chunks x 4 K-steps ----
    v2f aw[4];
#pragma unroll
    for (int ks = 0; ks < 4; ++ks)
      aw[ks] = *(const v2f*)(wlds + row16 * 16 + ks * 4 + kh2);
#pragma unroll
    for (int n = 0; n < 8; ++n) {
      v8f c = acc[n];
#pragma unroll
      for (int ks = 0; ks < 4; ++ks) {
        const int K0 = ks * 4 + kh2;
        v2f bf;
        bf.x = vlds[K0 * DDIM + n * 16 + row16];
        bf.y = vlds[(K0 + 1) * DDIM + n * 16 + row16];
        c = __builtin_amdgcn_wmma_f32_16x16x4_f32(false, aw[ks], false, bf,
                                                  (short)0, c, false, false);
      }
      acc[n] = c;
    }
  }

  // ---- emit partial: [m0..3, l0..3, acc[4][128]] ----
  float* p = part + ((size_t)((b * KVH + kvh) * NPART + chunk * NWAVES + wave))
                    * PSTRIDE;
  if (lane == 0) {
#pragma unroll
    for (int g = 0; g < GRP; ++g) { p[g] = mrow[g]; p[4 + g] = lrow[g]; }
  }
  if (lane < 16) {
#pragma unroll
    for (int g = 0; g < GRP; ++g)
#pragma unroll
      for (int n = 0; n < 8; ++n)
        p[8 + g * DDIM + n * 16 + lane] = acc[n][g];
  }
}

// ---------------- kernel 3: log-sum-exp merge of partials ----------------
__global__ void attn_reduce_kernel(const float* __restrict__ part,
                                   float* __restrict__ out) {
  const int b = blockIdx.x >> 3, kvh = blockIdx.x & 7;
  const int d = threadIdx.x;                       // 128 threads
  const float* pb = part + (size_t)((b * KVH + kvh) * NPART) * PSTRIDE;
#pragma unroll
  for (int g = 0; g < GRP; ++g) {
    float M = -__builtin_inff();
    for (int pp = 0; pp < NPART; ++pp)
      M = fmaxf(M, pb[pp * PSTRIDE + g]);
    float L = 0.f, Y = 0.f;
    for (int pp = 0; pp < NPART; ++pp) {
      const float w = __expf(pb[pp * PSTRIDE + g] - M);
      L += w * pb[pp * PSTRIDE + 4 + g];
      Y += w * pb[pp * PSTRIDE + 8 + g * DDIM + d];
    }
    out[((b * HDIM + g * KVH + kvh) << 7) + d] = Y / L;
  }
}

// ---------------- host launcher ----------------
extern "C" void kernel_launch(void* const* d_in, const int* in_sizes, int n_in,
                              void* d_out, int out_size, void* d_ws,
                              size_t ws_size, hipStream_t stream) {
  const float* Q    = (const float*)d_in[0];
  const float* K    = (const float*)d_in[1];
  const float* V    = (const float*)d_in[2];
  const float* Kc   = (const float*)d_in[3];
  const float* Vc   = (const float*)d_in[4];
  const float* cosb = (const float*)d_in[5];
  const float* sinb = (const float*)d_in[6];
  const int*   lens = (const int*)d_in[7];
  const float* mask = (const float*)d_in[8];
  // d_in[9] save_slots unused in decode path
  const int*   btab = (const int*)d_in[10];

  float* ws  = (float*)d_ws;
  float* qr  = ws;                              // 32768 f
  float* kq  = ws + BDIM * HDIM * DDIM;         // 8192 f
  float* prt = kq + BDIM * KVH * DDIM;          // 64*32*520 f (~4.1 MB)

  rope_kernel<<<(BDIM * (HDIM + KVH) * DDIM + 255) / 256, 256, 0, stream>>>(
      Q, K, cosb, sinb, qr, kq);

  attn_partial_kernel<<<dim3(NCHUNK, KVH, BDIM), NWAVES * 32, SMEM_B, stream>>>(
      qr, kq, V, Kc, Vc, lens, mask, btab, prt);

  attn_reduce_kernel<<<BDIM * KVH, DDIM, 0, stream>>>(prt, (float*)d_out);
}